// GAT_30820685316590
// MI455X (gfx1250) — compile-verified
//
#include <hip/hip_runtime.h>
#include <cstdint>

#define NN    50000
#define NEDGE 800000
#define DD    128
#define LEAKY 0.2f
#define EPSV  1e-8f

typedef __attribute__((ext_vector_type(2))) float v2f;
typedef __attribute__((ext_vector_type(8))) float v8f;

// Order-preserving float <-> uint mapping for atomicMax over signed floats.
__device__ __forceinline__ unsigned f2ord(float f) {
    unsigned u = __float_as_uint(f);
    return (f >= 0.0f) ? (u | 0x80000000u) : ~u;
}
__device__ __forceinline__ float ord2f(unsigned u) {
    return (u & 0x80000000u) ? __uint_as_float(u & 0x7FFFFFFFu) : __uint_as_float(~u);
}

// ---------------- init: zero out + den, reset max slot ----------------
__global__ void k_init(float* __restrict__ out, float* __restrict__ den,
                       unsigned* __restrict__ maxslot, int out_elems) {
    const int idx = blockIdx.x * blockDim.x + threadIdx.x;
    if (idx < out_elems) out[idx] = 0.0f;
    if (idx < NN)        den[idx] = 0.0f;
    if (idx == 0)        *maxslot = 0u;   // maps below every real float
}

// ---------------- H = V @ W_f via V_WMMA_F32_16X16X4_F32 ----------------
// One wave computes a 16x16 tile of H; 8 waves/block cover all 128 cols of a
// 16-row stripe. grid = N/16 = 3125 blocks.
__global__ void k_gemm(const float* __restrict__ V, const float* __restrict__ Wf,
                       float* __restrict__ H) {
    const int m0   = blockIdx.x << 4;
    const int wave = threadIdx.x >> 5;
    const int lane = threadIdx.x & 31;
    const int n0   = wave << 4;
    const int hi   = lane >> 4;      // lane group: K pairs 0/1 vs 2/3
    const int l15  = lane & 15;

    const float* arow = V  + (size_t)(m0 + l15) * DD + (hi << 1);
    const float* bcol = Wf + (size_t)(hi << 1) * DD + n0 + l15;

    v8f acc = {};
    #pragma unroll
    for (int k = 0; k < DD; k += 4) {
        v2f a = *(const v2f*)(arow + k);          // V[m0+l15][k+2*hi .. +1]
        v2f b;
        b.x = bcol[(size_t)k * DD];               // Wf[k+2*hi    ][n0+l15]
        b.y = bcol[(size_t)(k + 1) * DD];         // Wf[k+2*hi + 1][n0+l15]
        acc = __builtin_amdgcn_wmma_f32_16x16x4_f32(
            /*neg_a=*/false, a, /*neg_b=*/false, b,
            /*c_mod=*/(short)0, acc, /*reuse_a=*/false, /*reuse_b=*/false);
    }

    float* hbase = H + (size_t)(m0 + (hi << 3)) * DD + n0 + l15;
    #pragma unroll
    for (int r = 0; r < 8; ++r) hbase[(size_t)r * DD] = acc[r];
}

// ---------------- per-node scalars a_s = H.Watt[0:128], a_r = H.Watt[128:256]
__global__ void k_node_scal(const float* __restrict__ H, const float* __restrict__ Watt,
                            float* __restrict__ a_s, float* __restrict__ a_r) {
    const int wave = threadIdx.x >> 5;
    const int lane = threadIdx.x & 31;
    const int n = blockIdx.x * 8 + wave;
    if (n >= NN) return;
    const float4 h  = *(const float4*)(H + (size_t)n * DD + lane * 4);
    const float4 w1 = *(const float4*)(Watt + lane * 4);
    const float4 w2 = *(const float4*)(Watt + DD + lane * 4);
    float ds = h.x * w1.x + h.y * w1.y + h.z * w1.z + h.w * w1.w;
    float dr = h.x * w2.x + h.y * w2.y + h.z * w2.z + h.w * w2.w;
    #pragma unroll
    for (int off = 16; off > 0; off >>= 1) {
        ds += __shfl_xor(ds, off, 32);
        dr += __shfl_xor(dr, off, 32);
    }
    if (lane == 0) { a_s[n] = ds; a_r[n] = dr; }
}

// ---------------- per-edge logit + leaky-relu + global max ----------------
// One wave per edge (streams the 410MB E array once, coalesced 512B rows).
__global__ void k_edge_att(const float* __restrict__ E, const long long* __restrict__ edges,
                           const float* __restrict__ Watt, const float* __restrict__ b_att,
                           const float* __restrict__ a_s, const float* __restrict__ a_r,
                           float* __restrict__ att, unsigned* __restrict__ maxslot) {
    const int wave = threadIdx.x >> 5;
    const int lane = threadIdx.x & 31;
    const int e = blockIdx.x * 8 + wave;
    __shared__ float smax[8];
    float av = -3.4e38f;
    if (e < NEDGE) {
        const float4 ev = *(const float4*)(E + (size_t)e * DD + lane * 4);
        const float4 w3 = *(const float4*)(Watt + 2 * DD + lane * 4);
        float d = ev.x * w3.x + ev.y * w3.y + ev.z * w3.z + ev.w * w3.w;
        #pragma unroll
        for (int off = 16; off > 0; off >>= 1) d += __shfl_xor(d, off, 32);
        if (lane == 0) {
            const long long s = edges[2 * (size_t)e];
            const long long r = edges[2 * (size_t)e + 1];
            float a = a_s[s] + a_r[r] + d + b_att[0];
            a = (a > 0.0f) ? a : LEAKY * a;     // leaky_relu(0.2)
            att[e] = a;
            av = a;
        }
    }
    if (lane == 0) smax[wave] = av;
    __syncthreads();
    if (threadIdx.x == 0) {
        float m = smax[0];
        #pragma unroll
        for (int i = 1; i < 8; ++i) m = fmaxf(m, smax[i]);
        atomicMax(maxslot, f2ord(m));           // 1 atomic / 8 edges
    }
}

// ---------------- w = exp(att - max); scatter num/den by source node -------
// H / out / den all fit in 192MB L2, so gathers + fp32 atomics stay on-chip.
__global__ void k_scatter(const float* __restrict__ H, const long long* __restrict__ edges,
                          const float* __restrict__ att, const unsigned* __restrict__ maxslot,
                          float* __restrict__ out, float* __restrict__ den) {
    const int wave = threadIdx.x >> 5;
    const int lane = threadIdx.x & 31;
    const int e = blockIdx.x * 8 + wave;
    if (e >= NEDGE) return;
    const float gmax = ord2f(*maxslot);
    const float w = __expf(att[e] - gmax);
    const long long s = edges[2 * (size_t)e];
    const float4 h = *(const float4*)(H + (size_t)s * DD + lane * 4);
    float* o = out + (size_t)s * DD + lane * 4;
    unsafeAtomicAdd(o + 0, w * h.x);            // global_atomic_add_f32
    unsafeAtomicAdd(o + 1, w * h.y);
    unsafeAtomicAdd(o + 2, w * h.z);
    unsafeAtomicAdd(o + 3, w * h.w);
    if (lane == 0) unsafeAtomicAdd(den + s, w);
}

// ---------------- out = num / (den + eps) ----------------
__global__ void k_div(float* __restrict__ out, const float* __restrict__ den) {
    const int idx = blockIdx.x * blockDim.x + threadIdx.x;
    if (idx >= NN * DD) return;
    out[idx] = out[idx] / (den[idx >> 7] + EPSV);
}

extern "C" void kernel_launch(void* const* d_in, const int* in_sizes, int n_in,
                              void* d_out, int out_size, void* d_ws, size_t ws_size,
                              hipStream_t stream) {
    const float*     V     = (const float*)d_in[0];
    const float*     E     = (const float*)d_in[1];
    const long long* edges = (const long long*)d_in[2];   // int64 in reference
    const float*     Wf    = (const float*)d_in[3];
    const float*     Watt  = (const float*)d_in[4];
    const float*     b_att = (const float*)d_in[5];
    float* out = (float*)d_out;

    // workspace layout (floats): H[N*D] | a_s[N] | a_r[N] | den[N] | att[NE] | maxslot
    float* ws   = (float*)d_ws;
    float* H    = ws;
    float* a_s  = H + (size_t)NN * DD;
    float* a_r  = a_s + NN;
    float* den  = a_r + NN;
    float* att  = den + NN;
    unsigned* maxslot = (unsigned*)(att + NEDGE);

    k_init     <<<(NN * DD + 255) / 256, 256, 0, stream>>>(out, den, maxslot, NN * DD);
    k_gemm     <<<NN / 16,               256, 0, stream>>>(V, Wf, H);
    k_node_scal<<<(NN + 7) / 8,          256, 0, stream>>>(H, Watt, a_s, a_r);
    k_edge_att <<<(NEDGE + 7) / 8,       256, 0, stream>>>(E, edges, Watt, b_att, a_s, a_r, att, maxslot);
    k_scatter  <<<(NEDGE + 7) / 8,       256, 0, stream>>>(H, edges, att, maxslot, out, den);
    k_div      <<<(NN * DD + 255) / 256, 256, 0, stream>>>(out, den);
}